// GCNConv_diag_1580547974548
// MI455X (gfx1250) — compile-verified
//
#include <hip/hip_runtime.h>

typedef __attribute__((ext_vector_type(2))) float v2f;
typedef __attribute__((ext_vector_type(8))) float v8f;

#define BM 128
#define BN 128
#define BK 32
#define LDA 36   // BK + 4 pad: conflict-free b64 A-fragment reads, keeps 16B alignment
#define LDB 132  // BN + 4 pad

__global__ __launch_bounds__(256)
void gcn_diag_gemm(const float* __restrict__ X,   // input [N,D]
                   const float* __restrict__ A,   // [N,N]
                   const float* __restrict__ W,   // [D]
                   float* __restrict__ Out,       // [N,D]
                   int N, int D)
{
    __shared__ alignas(16) float As[2][BM * LDA];  // m-major: As[m*LDA + k]
    __shared__ alignas(16) float Bs[2][BK * LDB];  // k-major: Bs[k*LDB + n]

    const int tid  = threadIdx.x;
    const int lane = tid & 31;
    const int wave = tid >> 5;
    const int wm   = (wave >> 2) * 64;   // wave tile origin (M) in block tile
    const int wn   = (wave & 3) * 32;    // wave tile origin (N)

    const int m0 = blockIdx.y * BM;      // block row in A / Out
    const int n0 = blockIdx.x * BN;      // block col in X / Out

    // ---- global tile load indexing (fully coalesced float4) ----
    const int ar = tid >> 3;             // A: row in tile (0..31, x4 passes)
    const int ac = (tid & 7) * 4;        // A: col in tile
    const int br = tid >> 5;             // B: row in tile (0..7, x4 passes)
    const int bc = (tid & 31) * 4;       // B: col in tile

    // per-thread W scale: fixed columns across the whole K loop
    const float4 w4 = *(const float4*)&W[n0 + bc];

    v8f acc[4][2];
#pragma unroll
    for (int mi = 0; mi < 4; ++mi)
#pragma unroll
        for (int ni = 0; ni < 2; ++ni)
            acc[mi][ni] = v8f{};

    const int ktiles = N / BK;

    float4 aReg[4];
    float4 bReg[4];

    auto loadGlobal = [&](int kt) {
        const float* Ag = A + (size_t)m0 * N + (size_t)kt * BK;
#pragma unroll
        for (int p = 0; p < 4; ++p)
            aReg[p] = *(const float4*)&Ag[(size_t)(ar + 32 * p) * N + ac];
        const float* Xg = X + (size_t)kt * BK * D + n0;
#pragma unroll
        for (int p = 0; p < 4; ++p) {
            float4 v = *(const float4*)&Xg[(size_t)(br + 8 * p) * D + bc];
            v.x *= w4.x; v.y *= w4.y; v.z *= w4.z; v.w *= w4.w;
            bReg[p] = v;
        }
    };

    auto storeLds = [&](int buf) {
#pragma unroll
        for (int p = 0; p < 4; ++p) {
            *(float4*)&As[buf][(ar + 32 * p) * LDA + ac] = aReg[p];
            *(float4*)&Bs[buf][(br + 8 * p) * LDB + bc]  = bReg[p];
        }
    };

    const int ml = lane & 15;            // fragment M / N index within 16
    const int kk = (lane >> 4) * 2;      // lanes 16..31 hold K+2 / K+3

    auto compute = [&](int buf) {
#pragma unroll
        for (int k4 = 0; k4 < BK; k4 += 4) {
            v2f af[4], bf[2];
#pragma unroll
            for (int mi = 0; mi < 4; ++mi) {
                const float* s = &As[buf][(wm + mi * 16 + ml) * LDA + k4 + kk];
                af[mi].x = s[0];
                af[mi].y = s[1];
            }
#pragma unroll
            for (int ni = 0; ni < 2; ++ni) {
                bf[ni].x = Bs[buf][(k4 + kk)     * LDB + wn + ni * 16 + ml];
                bf[ni].y = Bs[buf][(k4 + kk + 1) * LDB + wn + ni * 16 + ml];
            }
#pragma unroll
            for (int mi = 0; mi < 4; ++mi)
#pragma unroll
                for (int ni = 0; ni < 2; ++ni)
                    acc[mi][ni] = __builtin_amdgcn_wmma_f32_16x16x4_f32(
                        false, af[mi], false, bf[ni],
                        (short)0, acc[mi][ni], false, false);
        }
    };

    // ---- pipelined main loop: prefetch tile kt+1 while computing tile kt ----
    loadGlobal(0);
    storeLds(0);
    __syncthreads();

    for (int kt = 0; kt < ktiles; ++kt) {
        const int buf = kt & 1;
        if (kt + 1 < ktiles) loadGlobal(kt + 1);
        compute(buf);
        if (kt + 1 < ktiles) storeLds(buf ^ 1);
        __syncthreads();
    }

    // ---- epilogue: C layout (VGPR v -> row v or v+8 per lane half) ----
#pragma unroll
    for (int mi = 0; mi < 4; ++mi)
#pragma unroll
        for (int ni = 0; ni < 2; ++ni) {
            const int col   = n0 + wn + ni * 16 + ml;
            const int rbase = m0 + wm + mi * 16 + (lane >> 4) * 8;
#pragma unroll
            for (int v = 0; v < 8; ++v)
                Out[(size_t)(rbase + v) * D + col] = acc[mi][ni][v];
        }
}

extern "C" void kernel_launch(void* const* d_in, const int* in_sizes, int n_in,
                              void* d_out, int out_size, void* d_ws, size_t ws_size,
                              hipStream_t stream) {
    const float* X = (const float*)d_in[0];   // input [N,D]
    const float* A = (const float*)d_in[1];   // A [N,N]
    const float* W = (const float*)d_in[2];   // W [D]
    float* Out = (float*)d_out;

    const int D = in_sizes[2];                // 512
    const int N = in_sizes[0] / D;            // 8192

    dim3 grid(D / BN, N / BM);                // (4, 64)
    gcn_diag_gemm<<<grid, 256, 0, stream>>>(X, A, W, Out, N, D);
}